// GGNNMsgPass_15960098472603
// MI455X (gfx1250) — compile-verified
//
#include <hip/hip_runtime.h>
#include <hip/hip_bf16.h>

// GGNN message passing, restructured:
//   s[b,i,e,n] = sum_{j: bfm[b,i,j]==e} afm[b,j,n]          (masked aggregation)
//   msg[b,i,m] = sum_{e,n} adj_w[e,m,n] * s[b,i,e,n]        (GEMM M=1024,K=320,N=64)
//   out = msg * att(a_bfm) + bias
// GEMM runs on V_WMMA_F32_16X16X4_F32 (fp32-exact vs reference up to reduction order).
// v2: dual independent WMMA accumulators (break the 80-deep C RAW chain),
//     branchless attention epilogue (no exec-mask divergence).

#define NF 64
#define EF 5
#define MF 64
#define Bb 16
#define Nn 64
#define KDIM (EF * NF)   // 320
#define ROWS (Bb * Nn)   // 1024

typedef float v2f __attribute__((ext_vector_type(2)));
typedef float v4f __attribute__((ext_vector_type(4)));
typedef float v8f __attribute__((ext_vector_type(8)));

// ---------------------------------------------------------------------------
// Kernel A: repack adj_w (EF, MF, NF) -> Wtp in "K-pair" layout:
//   Wtp[(k>>1)*128 + m*2 + (k&1)] = adj_w[e, m, n]   with k = e*64 + n
// so a WMMA B-operand lane reads its two consecutive K values as one b64 load.
// ---------------------------------------------------------------------------
__global__ __launch_bounds__(256) void ggnn_pack_w(const float* __restrict__ adj_w,
                                                   float* __restrict__ Wtp) {
    int idx = blockIdx.x * 256 + threadIdx.x;     // 0 .. 20479
    if (idx >= KDIM * MF) return;
    int k = idx >> 6;          // 0..319  (e*64 + n)
    int m = idx & 63;
    int e = k >> 6;
    int n = k & 63;
    float v = adj_w[e * (MF * NF) + m * NF + n];
    Wtp[(k >> 1) * (2 * MF) + m * 2 + (k & 1)] = v;
}

// ---------------------------------------------------------------------------
// Kernel B: fused aggregation + WMMA GEMM + attention/bias epilogue.
// One block = 16 consecutive rows r (r = b*64 + i), 128 threads = 4 waves.
// ---------------------------------------------------------------------------
__global__ __launch_bounds__(128) void ggnn_main(const float* __restrict__ afm,
                                                 const int*   __restrict__ bfm,
                                                 const int*   __restrict__ a_bfm,
                                                 const float* __restrict__ adj_a,
                                                 const float* __restrict__ bias,
                                                 const float* __restrict__ Wtp,
                                                 float*       __restrict__ out) {
    // s tile in K-pair layout: element (k,row) at (k>>1)*32 + row*2 + (k&1)
    __shared__ __align__(16) float s_lds[KDIM * 16];   // 20 KB

    const int tid = threadIdx.x;
    const int r0  = blockIdx.x * 16;    // first global row of this tile
    const int b   = r0 >> 6;            // all 16 rows share the same batch b
    const int i0  = r0 & 63;

    // ---------------- Phase 1: masked aggregation into s ----------------
    {
        const int team = tid >> 3;          // 0..15 : row within tile
        const int ln   = tid & 7;           // 0..7
        const int n0   = ln * 8;            // 8 features per thread
        const int i    = i0 + team;

        v4f acc0[EF], acc1[EF];
#pragma unroll
        for (int e = 0; e < EF; ++e) { acc0[e] = (v4f)0.0f; acc1[e] = (v4f)0.0f; }

        const float* arow = afm + (size_t)b * (Nn * NF);
        const int*   brow = bfm + (size_t)b * (Nn * Nn) + (size_t)i * Nn;

        for (int j = 0; j < Nn; ++j) {
            int et = brow[j];                                   // 0 = no edge
            v4f va = *(const v4f*)(arow + j * NF + n0);
            v4f vb = *(const v4f*)(arow + j * NF + n0 + 4);
#pragma unroll
            for (int e = 0; e < EF; ++e) {
                if (et == e + 1) { acc0[e] += va; acc1[e] += vb; }
            }
        }

        // write to LDS in K-pair layout (pairs of consecutive n are contiguous)
#pragma unroll
        for (int e = 0; e < EF; ++e) {
#pragma unroll
            for (int p = 0; p < 4; ++p) {
                int k = e * NF + n0 + p * 2;                    // even
                v2f pr;
                if      (p == 0) { pr[0] = acc0[e][0]; pr[1] = acc0[e][1]; }
                else if (p == 1) { pr[0] = acc0[e][2]; pr[1] = acc0[e][3]; }
                else if (p == 2) { pr[0] = acc1[e][0]; pr[1] = acc1[e][1]; }
                else             { pr[0] = acc1[e][2]; pr[1] = acc1[e][3]; }
                *(v2f*)&s_lds[(k >> 1) * 32 + team * 2] = pr;
            }
        }
    }
    __syncthreads();

    // ---------------- Phase 2: WMMA GEMM (16 x 320) x (320 x 64) ----------------
    // wave w owns output m-tile [w*16, w*16+16).
    const int wave = tid >> 5;           // 0..3
    const int lane = tid & 31;
    const int half = lane >> 4;          // 0: K-offsets {0,1}; 1: K-offsets {2,3}
    const int col  = lane & 15;          // A-row within tile AND B-column within tile
    const int m    = wave * 16 + col;

    // Two independent accumulators: even/odd K-steps. Linearity of D = A*B + C
    // makes the final elementwise merge exact.
    v8f cA = {};
    v8f cB = {};
#pragma unroll 2
    for (int k0 = 0; k0 < KDIM; k0 += 8) {
        {
            const int kk = k0 + half * 2;
            v2f av = *(const v2f*)&s_lds[(kk >> 1) * 32 + col * 2];
            v2f bv = *(const v2f*)&Wtp[(kk >> 1) * (2 * MF) + m * 2];
            cA = __builtin_amdgcn_wmma_f32_16x16x4_f32(
                     false, av, false, bv, (short)0, cA, false, false);
        }
        {
            const int kk = k0 + 4 + half * 2;
            v2f av = *(const v2f*)&s_lds[(kk >> 1) * 32 + col * 2];
            v2f bv = *(const v2f*)&Wtp[(kk >> 1) * (2 * MF) + m * 2];
            cB = __builtin_amdgcn_wmma_f32_16x16x4_f32(
                     false, av, false, bv, (short)0, cB, false, false);
        }
    }

    // ---------------- Epilogue: attention scale + bias (branchless) ----------------
    const float biasm = bias[m];
#pragma unroll
    for (int v = 0; v < 8; ++v) {
        // C layout: VGPR v, lanes 0-15 -> M=v ; lanes 16-31 -> M=v+8 ; N = lane&15
        int rowT = v + half * 8;
        int r    = r0 + rowT;                      // global row = b*64 + i
        int at   = a_bfm[r];                       // 0 = no attention -> zero
        int aidx = (at > 0 ? at : 1) - 1;          // clamp: always-valid gather
        float attv = adj_a[aidx * MF + m];         // unconditional load, no exec dance
        float att  = (at > 0) ? attv : 0.0f;       // v_cndmask, not a branch
        float msg  = cA[v] + cB[v];
        out[(size_t)r * MF + m] = msg * att + biasm;
    }
}

// ---------------------------------------------------------------------------
extern "C" void kernel_launch(void* const* d_in, const int* in_sizes, int n_in,
                              void* d_out, int out_size, void* d_ws, size_t ws_size,
                              hipStream_t stream) {
    const float* afm    = (const float*)d_in[0];   // (B, N, NF) f32
    const int*   bfm    = (const int*)  d_in[1];   // (B, N, N)  i32
    const int*   a_bfm  = (const int*)  d_in[2];   // (B, N)     i32
    const float* adj_w  = (const float*)d_in[3];   // (EF, MF, NF) f32
    const float* adj_a  = (const float*)d_in[4];   // (AEF, MF)  f32
    const float* bias   = (const float*)d_in[5];   // (MF,)      f32
    float*       out    = (float*)d_out;           // (B, N, MF) f32
    float*       Wtp    = (float*)d_ws;            // 320*64 floats = 80 KB scratch

    ggnn_pack_w<<<(KDIM * MF + 255) / 256, 256, 0, stream>>>(adj_w, Wtp);
    ggnn_main<<<ROWS / 16, 128, 0, stream>>>(afm, bfm, a_bfm, adj_a, bias, Wtp, out);
}